// QueryAndGroup_5669356832940
// MI455X (gfx1250) — compile-verified
//
#include <hip/hip_runtime.h>
#include <hip/hip_bf16.h>
#include <stdint.h>

// ---------------------------------------------------------------------------
// QueryAndGroup for MI455X (gfx1250, wave32)
//   Kernel 1: ball query. Squared distances computed on the matrix core via
//             V_WMMA_F32_16X16X4_F32 using d2 = |q|^2 - 2 q.p + |p|^2:
//               A(16x4)  row m : (-2qx, -2qy, -2qz, 1)
//               B(4x16)  col n : ( px ,  py ,  pz , |p|^2)
//               C(16x16)       : |q_m|^2 broadcast along the row
//             xyz tiles are staged into LDS with CDNA5 async-to-LDS copies
//             (double buffered, ASYNCcnt-tracked).
//   Kernel 2: 67-channel gather (bandwidth bound, ~70MB stores; feature table
//             is L2-resident on the 192MB L2 so gathers are cheap).
// ---------------------------------------------------------------------------

typedef __attribute__((ext_vector_type(2))) float v2f;
typedef __attribute__((ext_vector_type(8))) float v8f;
typedef __attribute__((ext_vector_type(4))) int   v4i;

#define B_   4
#define N_   16384
#define M_   2048
#define C_   64
#define S_   32            // nsample
#define CO_  (C_ + 3)
#define R2_  0.04f         // radius^2

#define TILE     128       // xyz points staged per LDS tile
#define WAVES    4         // waves per block (wave32)
#define QPW      16        // query rows per wave (one 16x16 WMMA tile)
#define THREADS1 (WAVES * 32)
#define NT       (N_ / TILE)

#define AS1 __attribute__((address_space(1)))
#define AS3 __attribute__((address_space(3)))

#if __has_builtin(__builtin_amdgcn_global_load_async_to_lds_b128) && \
    __has_builtin(__builtin_amdgcn_s_wait_asynccnt)
#define HAVE_ASYNC 1
#else
#define HAVE_ASYNC 0
#endif

__device__ __forceinline__ unsigned ballot32(bool p) {
#if __has_builtin(__builtin_amdgcn_ballot_w32)
  return __builtin_amdgcn_ballot_w32(p);
#else
  return (unsigned)__ballot((int)p);
#endif
}

// ---------------------------------------------------------------------------
// Kernel 1: ball query via WMMA distance tiles.
// Each wave owns 16 query points; the block shares LDS xyz tiles.
// ---------------------------------------------------------------------------
__global__ __launch_bounds__(THREADS1)
void ball_query_wmma(const float* __restrict__ xyz,
                     const float* __restrict__ new_xyz,
                     int* __restrict__ idx_out) {
  __shared__ float s_xyz[2][TILE * 3];
  __shared__ float s_qn2[WAVES * QPW];

  const int tid  = threadIdx.x;
  const int lane = tid & 31;
  const int wave = tid >> 5;
  const int hi   = lane >> 4;      // 0: lanes 0-15, 1: lanes 16-31
  const int t    = lane & 15;
  const int b    = blockIdx.y;
  const int mbase = blockIdx.x * (WAVES * QPW) + wave * QPW;
  const int mq    = mbase + t;     // query row owned by this lane (dup on hi half)

  // Query coordinates (lanes n and n+16 read the same row -> broadcastable).
  const size_t qoff = ((size_t)b * M_ + mq) * 3;
  const float qx = new_xyz[qoff + 0];
  const float qy = new_xyz[qoff + 1];
  const float qz = new_xyz[qoff + 2];
  if (!hi) s_qn2[wave * QPW + t] = qx * qx + qy * qy + qz * qz;

  // A operand, 16x4 f32 layout: VGPR0 = {K0 | K2}, VGPR1 = {K1 | K3}.
  v2f amat;
  amat.x = hi ? (-2.0f * qz) : (-2.0f * qx);   // K2 | K0
  amat.y = hi ? 1.0f         : (-2.0f * qy);   // K3 | K1

  __syncthreads();

  // C operand: C[m][n] = |q_m|^2. VGPR r holds rows r (lanes 0-15) and r+8.
  v8f cmat;
#pragma unroll
  for (int r = 0; r < 8; ++r)
    cmat[r] = s_qn2[wave * QPW + r + (hi ? 8 : 0)];

  int cnt = 0;
  int first_idx = 0;
  const size_t row_off = ((size_t)b * M_ + mq) * S_;
  const float* gxyz = xyz + (size_t)b * N_ * 3;

#if HAVE_ASYNC
  auto issue_tile = [&](int buf, int tile) {
    if (tid < (TILE * 3) / 4) {                       // 96 x 16B per tile
      const float* g = gxyz + (size_t)tile * TILE * 3 + tid * 4;
      float* l = &s_xyz[buf][tid * 4];
      __builtin_amdgcn_global_load_async_to_lds_b128((AS1 v4i*)g,
                                                     (AS3 v4i*)l, 0, 0);
    }
  };
#endif

  auto process_tile = [&](int buf, int tile) {
    // Wave-uniform early-out once all 16 rows are full (EXEC stays all-ones
    // around the WMMA as required by the ISA).
    if (ballot32((lane < QPW) && (cnt < S_)) == 0u) return;
#pragma unroll
    for (int sub = 0; sub < TILE / 16; ++sub) {
      const int nl = sub * 16 + t;
      const float px = s_xyz[buf][nl * 3 + 0];
      const float py = s_xyz[buf][nl * 3 + 1];
      const float pz = s_xyz[buf][nl * 3 + 2];
      const float pn2 = px * px + py * py + pz * pz;
      v2f bmat;                                  // B 4x16: VGPR0={K0|K2}, VGPR1={K1|K3}
      bmat.x = hi ? pz  : px;
      bmat.y = hi ? pn2 : py;
      // D[m][n] = |q_m|^2 - 2 q_m . p_n + |p_n|^2
      v8f d = __builtin_amdgcn_wmma_f32_16x16x4_f32(
          false, amat, false, bmat, (short)0, cmat, false, false);
      unsigned mymask = 0u;
#pragma unroll
      for (int r = 0; r < 8; ++r) {
        const unsigned bal = ballot32(d[r] < R2_);     // rows r | r+8
        if (lane == r)          mymask = bal & 0xffffu;
        else if (lane == r + 8) mymask = bal >> 16;
      }
      const int nbase = tile * TILE + sub * 16;
      while (mymask) {                                  // append in point order
        const int bit = __builtin_ctz(mymask);
        mymask &= mymask - 1;
        if (cnt < S_) {
          const int n = nbase + bit;
          if (cnt == 0) first_idx = n;
          idx_out[row_off + cnt] = n;
          ++cnt;
        }
      }
    }
  };

#if HAVE_ASYNC
  // Double-buffered async DMA pipeline: overlap LDS fill with WMMA scan.
  int cur = 0;
  issue_tile(0, 0);
  for (int tile = 0; tile < NT; ++tile) {
    __builtin_amdgcn_s_wait_asynccnt(0);
    __syncthreads();                        // tile data visible to all waves
    if (tile + 1 < NT) issue_tile(cur ^ 1, tile + 1);
    process_tile(cur, tile);
    cur ^= 1;
  }
#else
  for (int tile = 0; tile < NT; ++tile) {
    __syncthreads();
    for (int i = tid; i < TILE * 3; i += THREADS1)
      s_xyz[0][i] = gxyz[(size_t)tile * TILE * 3 + i];
    __syncthreads();
    process_tile(0, tile);
  }
#endif

  // Pad remaining slots with the first hit (0 if the ball is empty).
  if (lane < QPW) {
    const int fv = (cnt > 0) ? first_idx : 0;
    for (int s = cnt; s < S_; ++s) idx_out[row_off + s] = fv;
  }
}

// ---------------------------------------------------------------------------
// Kernel 2: gather. One thread per (b, channel, query) output row of 32.
// ---------------------------------------------------------------------------
__global__ __launch_bounds__(256)
void group_gather(const float* __restrict__ xyz,
                  const float* __restrict__ new_xyz,
                  const float* __restrict__ feat,
                  const int* __restrict__ idx,
                  float* __restrict__ out) {
  const int gtid = blockIdx.x * blockDim.x + threadIdx.x;
  if (gtid >= B_ * CO_ * M_) return;
  const int m = gtid % M_;
  const int c = (gtid / M_) % CO_;
  const int b = gtid / (M_ * CO_);

  const int* ip = idx + ((size_t)b * M_ + m) * S_;
  __builtin_prefetch(ip, 0, 0);              // global_prefetch_b8

  int ibuf[S_];
  const int4* ip4 = (const int4*)ip;         // 128B-aligned rows
#pragma unroll
  for (int v = 0; v < S_ / 4; ++v) {
    const int4 i4 = ip4[v];
    ibuf[4 * v + 0] = i4.x; ibuf[4 * v + 1] = i4.y;
    ibuf[4 * v + 2] = i4.z; ibuf[4 * v + 3] = i4.w;
  }

  float row[S_];
  if (c < 3) {
    const float q = new_xyz[((size_t)b * M_ + m) * 3 + c];
    const float* px = xyz + (size_t)b * N_ * 3;
#pragma unroll
    for (int s = 0; s < S_; ++s)
      row[s] = px[(size_t)ibuf[s] * 3 + c] - q;
  } else {
    const float* pf = feat + ((size_t)b * C_ + (c - 3)) * N_;
#pragma unroll
    for (int s = 0; s < S_; ++s)
      row[s] = pf[ibuf[s]];
  }

  float4* o4 = (float4*)(out + (size_t)gtid * S_);
#pragma unroll
  for (int v = 0; v < S_ / 4; ++v)
    o4[v] = make_float4(row[4 * v], row[4 * v + 1],
                        row[4 * v + 2], row[4 * v + 3]);
}

// ---------------------------------------------------------------------------
extern "C" void kernel_launch(void* const* d_in, const int* in_sizes, int n_in,
                              void* d_out, int out_size, void* d_ws, size_t ws_size,
                              hipStream_t stream) {
  (void)in_sizes; (void)n_in; (void)out_size; (void)ws_size;
  const float* xyz     = (const float*)d_in[0];   // (B,N,3)
  const float* new_xyz = (const float*)d_in[1];   // (B,M,3)
  const float* feat    = (const float*)d_in[2];   // (B,C,N)
  int*   idx = (int*)d_ws;                        // (B,M,S) scratch: 1 MB
  float* out = (float*)d_out;                     // (B,C+3,M,S)

  dim3 g1(M_ / (WAVES * QPW), B_);                // 32 x 4 blocks, 128 thr
  ball_query_wmma<<<g1, THREADS1, 0, stream>>>(xyz, new_xyz, idx);

  const int total = B_ * CO_ * M_;
  group_gather<<<(total + 255) / 256, 256, 0, stream>>>(xyz, new_xyz, feat,
                                                        idx, out);
}